// realignment_layer_6004364280021
// MI455X (gfx1250) — compile-verified
//
#include <hip/hip_runtime.h>
#include <hip/hip_bf16.h>
#include <math.h>

typedef __attribute__((ext_vector_type(2))) float v2f;
typedef __attribute__((ext_vector_type(4))) float v4f;
typedef __attribute__((ext_vector_type(8))) float v8f;

// Analytic rigid inverse of [R(q) t; 0 1]:  Rinv = R^T (row-major in R[9]),
// tinv = -R^T t.  Matches jnp.linalg.inv of the well-conditioned rigid
// transform to f32 roundoff.
__device__ __forceinline__ void make_rinv(const float* __restrict__ q,
                                          const float* __restrict__ dt,
                                          int b, float R[9], float ti[3]) {
    float qw = q[4 * b + 0], qx = q[4 * b + 1], qy = q[4 * b + 2], qz = q[4 * b + 3];
    float s = 1.0f / sqrtf(qw * qw + qx * qx + qy * qy + qz * qz);
    qw *= s; qx *= s; qy *= s; qz *= s;
    float R00 = 1.f - 2.f * (qy * qy + qz * qz);
    float R01 = 2.f * (qx * qy - qw * qz);
    float R02 = 2.f * (qx * qz + qw * qy);
    float R10 = 2.f * (qx * qy + qw * qz);
    float R11 = 1.f - 2.f * (qx * qx + qz * qz);
    float R12 = 2.f * (qy * qz - qw * qx);
    float R20 = 2.f * (qx * qz - qw * qy);
    float R21 = 2.f * (qy * qz + qw * qx);
    float R22 = 1.f - 2.f * (qx * qx + qy * qy);
    // Rinv = R^T, row-major
    R[0] = R00; R[1] = R10; R[2] = R20;
    R[3] = R01; R[4] = R11; R[5] = R21;
    R[6] = R02; R[7] = R12; R[8] = R22;
    float t0 = dt[3 * b + 0], t1 = dt[3 * b + 1], t2 = dt[3 * b + 2];
    ti[0] = -(R00 * t0 + R10 * t1 + R20 * t2);
    ti[1] = -(R01 * t0 + R11 * t1 + R21 * t2);
    ti[2] = -(R02 * t0 + R12 * t1 + R22 * t2);
}

// ---------------------------------------------------------------------------
// Setup kernel (B == 16), one wave32.  Stage 1: lanes 0-15 build the 4x4
// rigid inverses into LDS (and ws).  Stage 2: 4x V_WMMA_F32_16X16X4_F32, each
// computing 4 batches' (dT^-1 @ T_mis) as the diagonal 4x4 blocks of a 16x16
// f32 accumulator.
// ---------------------------------------------------------------------------
__global__ __launch_bounds__(32) void setup_wmma_kernel(
    const float* __restrict__ T, const float* __restrict__ q,
    const float* __restrict__ dt, float* __restrict__ outT,
    float* __restrict__ ws) {
    __shared__ float sA[256];  // [batch][i*4+k]  = dT^-1 of each batch
    int lane = threadIdx.x;
    if (lane < 16) {
        int b = lane;
        float R[9], ti[3];
        make_rinv(q, dt, b, R, ti);
        sA[b * 16 + 0]  = R[0]; sA[b * 16 + 1]  = R[1]; sA[b * 16 + 2]  = R[2]; sA[b * 16 + 3]  = ti[0];
        sA[b * 16 + 4]  = R[3]; sA[b * 16 + 5]  = R[4]; sA[b * 16 + 6]  = R[5]; sA[b * 16 + 7]  = ti[1];
        sA[b * 16 + 8]  = R[6]; sA[b * 16 + 9]  = R[7]; sA[b * 16 + 10] = R[8]; sA[b * 16 + 11] = ti[2];
        sA[b * 16 + 12] = 0.f;  sA[b * 16 + 13] = 0.f;  sA[b * 16 + 14] = 0.f;  sA[b * 16 + 15] = 1.f;
        if (ws) {
            #pragma unroll
            for (int i = 0; i < 9; ++i) ws[b * 12 + i] = R[i];
            ws[b * 12 + 9] = ti[0]; ws[b * 12 + 10] = ti[1]; ws[b * 12 + 11] = ti[2];
        }
    }
    __syncthreads();

    int r  = lane & 15;   // column of B / row of A held by this lane
    int hi = lane >> 4;   // half-wave selector per ISA A/B layouts
    int qb = r >> 2;      // which 4x4 block within the group of 4 batches
    int qi = r & 3;       // local row (A) / local col (B)
    #pragma unroll
    for (int g = 0; g < 4; ++g) {
        int batch = g * 4 + qb;
        // A(16x4): lane L vgpr v = A[L%16][2*(L/16)+v]
        v2f a;
        a.x = sA[batch * 16 + qi * 4 + 2 * hi + 0];
        a.y = sA[batch * 16 + qi * 4 + 2 * hi + 1];
        // B(4x16): lane L vgpr v = B[2*(L/16)+v][L%16]
        v2f bm;
        bm.x = T[batch * 16 + (2 * hi + 0) * 4 + qi];
        bm.y = T[batch * 16 + (2 * hi + 1) * 4 + qi];
        v8f c = {};
        c = __builtin_amdgcn_wmma_f32_16x16x4_f32(
            /*neg_a=*/false, a, /*neg_b=*/false, bm,
            /*c_mod=*/(short)0, c, /*reuse_a=*/false, /*reuse_b=*/false);
        // C(16x16): lane L vgpr v = C[v+8*(L/16)][L%16]; keep diagonal blocks.
        #pragma unroll
        for (int v = 0; v < 8; ++v) {
            int m = v + 8 * hi;
            if ((m >> 2) == qb) {
                outT[batch * 16 + (m & 3) * 4 + qi] = c[v];
            }
        }
    }
}

// Generic-B scalar fallback (not expected with this harness's B=16).
__global__ void setup_scalar_kernel(const float* __restrict__ T,
                                    const float* __restrict__ q,
                                    const float* __restrict__ dt,
                                    float* __restrict__ outT,
                                    float* __restrict__ ws, int B) {
    int b = blockIdx.x * blockDim.x + threadIdx.x;
    if (b >= B) return;
    float R[9], ti[3];
    make_rinv(q, dt, b, R, ti);
    float Ainv[16] = {R[0], R[1], R[2], ti[0],
                      R[3], R[4], R[5], ti[1],
                      R[6], R[7], R[8], ti[2],
                      0.f,  0.f,  0.f,  1.f};
    for (int i = 0; i < 4; ++i)
        for (int j = 0; j < 4; ++j) {
            float s = 0.f;
            for (int k = 0; k < 4; ++k) s += Ainv[i * 4 + k] * T[b * 16 + k * 4 + j];
            outT[b * 16 + i * 4 + j] = s;
        }
    if (ws) {
        for (int i = 0; i < 9; ++i) ws[b * 12 + i] = R[i];
        ws[b * 12 + 9] = ti[0]; ws[b * 12 + 10] = ti[1]; ws[b * 12 + 11] = ti[2];
    }
}

// ---------------------------------------------------------------------------
// Streaming kernel: bandwidth-bound.  4 points (48 B) per thread via three
// 128-bit non-temporal loads/stores (192 MB streamed once, no reuse -> NT
// keeps it out of L2's way).  Per-batch constants come from ws (scalarizable
// uniform loads) or are recomputed inline if ws is unavailable.
// ---------------------------------------------------------------------------
__global__ __launch_bounds__(256) void realign_kernel(
    const float* __restrict__ pcd, const float* __restrict__ q,
    const float* __restrict__ dt, const float* __restrict__ ws,
    float* __restrict__ outP, int N, int gpb) {
    int b = blockIdx.y;
    int g = blockIdx.x * blockDim.x + threadIdx.x;
    if (g >= gpb) return;

    float R[9], ti[3];
    if (ws) {
        const float* w = ws + b * 12;
        #pragma unroll
        for (int i = 0; i < 9; ++i) R[i] = w[i];
        ti[0] = w[9]; ti[1] = w[10]; ti[2] = w[11];
    } else {
        make_rinv(q, dt, b, R, ti);
    }

    long long p0 = (long long)g * 4;
    size_t base = (size_t)b * (size_t)N * 3 + (size_t)p0 * 3;

    if (p0 + 4 <= N) {
        const v4f* src = (const v4f*)(pcd + base);
        v4f va = __builtin_nontemporal_load(src + 0);
        v4f vb = __builtin_nontemporal_load(src + 1);
        v4f vc = __builtin_nontemporal_load(src + 2);
        float f[12] = {va.x, va.y, va.z, va.w,
                       vb.x, vb.y, vb.z, vb.w,
                       vc.x, vc.y, vc.z, vc.w};
        float o[12];
        #pragma unroll
        for (int p = 0; p < 4; ++p) {
            float px = f[p * 3 + 0], py = f[p * 3 + 1], pz = f[p * 3 + 2];
            o[p * 3 + 0] = fmaf(R[0], px, fmaf(R[1], py, fmaf(R[2], pz, ti[0])));
            o[p * 3 + 1] = fmaf(R[3], px, fmaf(R[4], py, fmaf(R[5], pz, ti[1])));
            o[p * 3 + 2] = fmaf(R[6], px, fmaf(R[7], py, fmaf(R[8], pz, ti[2])));
        }
        v4f o0 = {o[0], o[1], o[2], o[3]};
        v4f o1 = {o[4], o[5], o[6], o[7]};
        v4f o2 = {o[8], o[9], o[10], o[11]};
        v4f* dst = (v4f*)(outP + base);
        __builtin_nontemporal_store(o0, dst + 0);
        __builtin_nontemporal_store(o1, dst + 1);
        __builtin_nontemporal_store(o2, dst + 2);
    } else {
        // tail (N % 4 != 0)
        for (long long p = p0; p < N; ++p) {
            size_t idx = (size_t)b * (size_t)N * 3 + (size_t)p * 3;
            float px = pcd[idx + 0], py = pcd[idx + 1], pz = pcd[idx + 2];
            outP[idx + 0] = fmaf(R[0], px, fmaf(R[1], py, fmaf(R[2], pz, ti[0])));
            outP[idx + 1] = fmaf(R[3], px, fmaf(R[4], py, fmaf(R[5], pz, ti[1])));
            outP[idx + 2] = fmaf(R[6], px, fmaf(R[7], py, fmaf(R[8], pz, ti[2])));
        }
    }
}

extern "C" void kernel_launch(void* const* d_in, const int* in_sizes, int n_in,
                              void* d_out, int out_size, void* d_ws, size_t ws_size,
                              hipStream_t stream) {
    const float* pcd = (const float*)d_in[0];   // [B,N,3]
    const float* T   = (const float*)d_in[1];   // [B,4,4]
    const float* q   = (const float*)d_in[2];   // [B,4]
    const float* dt  = (const float*)d_in[3];   // [B,3]

    int B = in_sizes[2] / 4;
    long long pcd_elems = in_sizes[0];
    int N = (int)(pcd_elems / ((long long)B * 3));

    float* outT = (float*)d_out;                 // batch_T_pred: B*16 floats
    float* outP = outT + (size_t)B * 16;         // pcd_realigned: B*N*3 floats
    float* ws = (ws_size >= (size_t)B * 12 * sizeof(float)) ? (float*)d_ws : nullptr;

    if (B == 16) {
        setup_wmma_kernel<<<1, 32, 0, stream>>>(T, q, dt, outT, ws);
    } else {
        int thr = 64;
        setup_scalar_kernel<<<(B + thr - 1) / thr, thr, 0, stream>>>(T, q, dt, outT, ws, B);
    }

    int gpb = (N + 3) / 4;                       // 4-point groups per batch
    dim3 grid((unsigned)((gpb + 255) / 256), (unsigned)B);
    realign_kernel<<<grid, 256, 0, stream>>>(pcd, q, dt, ws, outP, N, gpb);
}